// KGIE_52106543235208
// MI455X (gfx1250) — compile-verified
//
#include <hip/hip_runtime.h>
#include <math.h>

#define BDIM 64   // embedding dim
#define KNBR 8    // neighbors per node

typedef float v2f __attribute__((ext_vector_type(2)));
typedef float v8f __attribute__((ext_vector_type(8)));

// ---------------------------------------------------------------------------
// 1) user_vec: usr_table[u] * rel_table[useradj] -> K-channel size-3 conv over D
//    (only the middle column of the 3x3 kernel contributes since W-dim == 1)
// ---------------------------------------------------------------------------
__global__ void user_vec_kernel(const int* __restrict__ u,
                                const int* __restrict__ useradj,
                                const float* __restrict__ usr_table,
                                const float* __restrict__ rel_table,
                                const float* __restrict__ conv_w,
                                const float* __restrict__ conv_b,
                                float* __restrict__ uv)
{
    __shared__ float x[KNBR][BDIM + 2];      // +1 halo each side
    int b = blockIdx.x;
    int d = threadIdx.x;                     // 0..63
    float us = usr_table[(size_t)u[b] * BDIM + d];
#pragma unroll
    for (int k = 0; k < KNBR; ++k) {
        int r = useradj[b * KNBR + k];
        x[k][d + 1] = us * rel_table[(size_t)r * BDIM + d];
        if (d == 0) { x[k][0] = 0.f; x[k][BDIM + 1] = 0.f; }
    }
    __syncthreads();
    float y = conv_b[0];
#pragma unroll
    for (int k = 0; k < KNBR; ++k)
#pragma unroll
        for (int j = 0; j < 3; ++j)
            y += x[k][d + j] * conv_w[k * 9 + j * 3 + 1];   // w[0,k,j,1] (OIHW)
    uv[(size_t)b * BDIM + d] = y;
}

// ---------------------------------------------------------------------------
// 2) two-hop index gather
// ---------------------------------------------------------------------------
__global__ void gather_kernel(const int* __restrict__ v,
                              const int* __restrict__ adj_ent,
                              const int* __restrict__ adj_rel,
                              int* __restrict__ ents1, int* __restrict__ rels0,
                              int* __restrict__ ents2, int* __restrict__ rels1)
{
    int b = blockIdx.x;
    int t = threadIdx.x;                     // 0..63
    int vb = v[b];
    if (t < KNBR) {
        ents1[b * KNBR + t] = adj_ent[(size_t)vb * KNBR + t];
        rels0[b * KNBR + t] = adj_rel[(size_t)vb * KNBR + t];
    }
    int e1 = adj_ent[(size_t)vb * KNBR + (t >> 3)];
    ents2[b * 64 + t] = adj_ent[(size_t)e1 * KNBR + (t & 7)];
    rels1[b * 64 + t] = adj_rel[(size_t)e1 * KNBR + (t & 7)];
}

// ---------------------------------------------------------------------------
// 3) aggregation: T[r,:] = self[r,:] + sum_k softmax_k(uv[b].rel_k) * nbr_k[:]
//    one 64-thread block per row; parameterized for all three stages
// ---------------------------------------------------------------------------
__global__ void agg_kernel(const float* __restrict__ uv,
                           const float* __restrict__ self_base,
                           const int*   __restrict__ self_idx,   // null -> identity
                           const float* __restrict__ nbr_base,
                           const int*   __restrict__ nbr_idx,    // null -> identity
                           const float* __restrict__ rel_table,
                           const int*   __restrict__ rel_idx,
                           float* __restrict__ T,
                           int bshift)
{
    __shared__ float prod[KNBR][BDIM];
    __shared__ float sw[KNBR];
    int r = blockIdx.x;
    int d = threadIdx.x;                     // 0..63
    int b = r >> bshift;
    float uvl = uv[(size_t)b * BDIM + d];
#pragma unroll
    for (int k = 0; k < KNBR; ++k) {
        int ri = rel_idx[r * KNBR + k];
        prod[k][d] = uvl * rel_table[(size_t)ri * BDIM + d];
    }
    __syncthreads();
    if (d < KNBR) {
        float s = 0.f;
        for (int j = 0; j < BDIM; ++j) s += prod[d][j];
        sw[d] = s;
    }
    __syncthreads();
    if (d == 0) {
        float m = sw[0];
        for (int k = 1; k < KNBR; ++k) m = fmaxf(m, sw[k]);
        float e[KNBR], tot = 0.f;
        for (int k = 0; k < KNBR; ++k) { e[k] = expf(sw[k] - m); tot += e[k]; }
        float inv = 1.f / tot;
        for (int k = 0; k < KNBR; ++k) sw[k] = e[k] * inv;
    }
    __syncthreads();
    int si = self_idx ? self_idx[r] : r;
    float acc = self_base[(size_t)si * BDIM + d];
#pragma unroll
    for (int k = 0; k < KNBR; ++k) {
        int ni = nbr_idx ? nbr_idx[r * KNBR + k] : (r * KNBR + k);
        acc += sw[k] * nbr_base[(size_t)ni * BDIM + d];
    }
    T[(size_t)r * BDIM + d] = acc;
}

// ---------------------------------------------------------------------------
// 4) Y = act(X @ W + bias), X:[rows,64], W:[64,64] -- CDNA5 WMMA f32 path.
//    One wave32 per 16-row tile; 4 N-tiles of 16; 16 K-steps of K=4:
//    64 x V_WMMA_F32_16X16X4_F32 per wave.
//    A 16x4 f32: lane holds M=lane%16; regs hold K = 2*(lane/16)+{0,1}
//    B 4x16 f32: lane holds N=lane%16; regs hold same K pair
//    C/D 16x16 f32: lane n+16*(m/8), vgpr m%8
// ---------------------------------------------------------------------------
__global__ void gemm64_wmma_kernel(const float* __restrict__ X,
                                   const float* __restrict__ W,
                                   const float* __restrict__ bias,
                                   float* __restrict__ Y,
                                   int rows, int act /*0=relu,1=tanh*/)
{
    int wave = (blockIdx.x * blockDim.x + threadIdx.x) >> 5;
    int lane = threadIdx.x & 31;
    int row0 = wave * 16;
    if (row0 >= rows) return;                // wave-uniform; EXEC stays all-1s
    int m    = lane & 15;
    int half = lane >> 4;

    v8f acc0 = {}, acc1 = {}, acc2 = {}, acc3 = {};
    const float* xr = X + (size_t)(row0 + m) * BDIM;
#pragma unroll
    for (int kk = 0; kk < 16; ++kk) {
        int kb = 4 * kk + 2 * half;
        v2f a;  a.x = xr[kb];  a.y = xr[kb + 1];
        const float* w0 = W + (size_t)kb * BDIM + m;
        const float* w1 = w0 + BDIM;
        v2f b0, b1, b2, b3;
        b0.x = w0[ 0]; b0.y = w1[ 0];
        b1.x = w0[16]; b1.y = w1[16];
        b2.x = w0[32]; b2.y = w1[32];
        b3.x = w0[48]; b3.y = w1[48];
        acc0 = __builtin_amdgcn_wmma_f32_16x16x4_f32(false, a, false, b0, (short)0, acc0, false, false);
        acc1 = __builtin_amdgcn_wmma_f32_16x16x4_f32(false, a, false, b1, (short)0, acc1, false, false);
        acc2 = __builtin_amdgcn_wmma_f32_16x16x4_f32(false, a, false, b2, (short)0, acc2, false, false);
        acc3 = __builtin_amdgcn_wmma_f32_16x16x4_f32(false, a, false, b3, (short)0, acc3, false, false);
    }

    v8f accs[4] = {acc0, acc1, acc2, acc3};
#pragma unroll
    for (int nt = 0; nt < 4; ++nt) {
#pragma unroll
        for (int j = 0; j < 8; ++j) {
            int mr = 8 * half + j;
            int n  = nt * 16 + m;
            float val = accs[nt][j] + bias[n];
            val = act ? tanhf(val) : fmaxf(val, 0.f);
            Y[(size_t)(row0 + mr) * BDIM + n] = val;
        }
    }
}

// ---------------------------------------------------------------------------
// 5) item-side conv + dot(user_vec, item_vec) + sigmoid
// ---------------------------------------------------------------------------
__global__ void final_kernel(const int* __restrict__ itemadj,
                             const float* __restrict__ usr_table,
                             const float* __restrict__ item,
                             const float* __restrict__ uv,
                             const float* __restrict__ conv_w,
                             const float* __restrict__ conv_b,
                             float* __restrict__ out)
{
    __shared__ float x[KNBR][BDIM + 2];
    __shared__ float red[BDIM];
    int b = blockIdx.x;
    int d = threadIdx.x;
    float it = item[(size_t)b * BDIM + d];
#pragma unroll
    for (int k = 0; k < KNBR; ++k) {
        int r = itemadj[b * KNBR + k];
        x[k][d + 1] = it * usr_table[(size_t)r * BDIM + d];
        if (d == 0) { x[k][0] = 0.f; x[k][BDIM + 1] = 0.f; }
    }
    __syncthreads();
    float y = conv_b[0];
#pragma unroll
    for (int k = 0; k < KNBR; ++k)
#pragma unroll
        for (int j = 0; j < 3; ++j)
            y += x[k][d + j] * conv_w[k * 9 + j * 3 + 1];
    red[d] = y * uv[(size_t)b * BDIM + d];
    for (int s = 32; s > 0; s >>= 1) {
        __syncthreads();
        if (d < s) red[d] += red[d + s];
    }
    if (d == 0) out[b] = 1.f / (1.f + expf(-red[0]));
}

// ---------------------------------------------------------------------------
// launch
// ---------------------------------------------------------------------------
extern "C" void kernel_launch(void* const* d_in, const int* in_sizes, int n_in,
                              void* d_out, int out_size, void* d_ws, size_t ws_size,
                              hipStream_t stream)
{
    (void)n_in; (void)out_size; (void)ws_size;
    const int*   u         = (const int*)  d_in[0];
    const int*   v         = (const int*)  d_in[1];
    const int*   useradj   = (const int*)  d_in[2];
    const int*   itemadj   = (const int*)  d_in[3];
    const int*   adj_ent   = (const int*)  d_in[4];
    const int*   adj_rel   = (const int*)  d_in[5];
    const float* usr_table = (const float*)d_in[6];
    const float* ent_table = (const float*)d_in[7];
    const float* rel_table = (const float*)d_in[8];
    const float* conv_w    = (const float*)d_in[9];
    const float* conv_b    = (const float*)d_in[10];
    const float* agg_w     = (const float*)d_in[11];
    const float* agg_b     = (const float*)d_in[12];
    float* out = (float*)d_out;

    const int B = in_sizes[0];

    // workspace carve-out
    char* ws = (char*)d_ws;
    size_t o = 0;
    float* uv    = (float*)(ws + o); o += (size_t)B * BDIM * 4;
    int*   ents1 = (int*)  (ws + o); o += (size_t)B * KNBR * 4;
    int*   rels0 = (int*)  (ws + o); o += (size_t)B * KNBR * 4;
    int*   ents2 = (int*)  (ws + o); o += (size_t)B * 64 * 4;
    int*   rels1 = (int*)  (ws + o); o += (size_t)B * 64 * 4;
    float* T1    = (float*)(ws + o); o += (size_t)B * KNBR * BDIM * 4;
    float* ev1   = (float*)(ws + o); o += (size_t)B * KNBR * BDIM * 4;
    float* T0    = (float*)(ws + o); o += (size_t)B * BDIM * 4;
    float* ev0   = (float*)(ws + o); o += (size_t)B * BDIM * 4;
    float* T2    = (float*)(ws + o); o += (size_t)B * BDIM * 4;
    float* item  = (float*)(ws + o); o += (size_t)B * BDIM * 4;

    // 1) user vector
    user_vec_kernel<<<B, BDIM, 0, stream>>>(u, useradj, usr_table, rel_table,
                                            conv_w, conv_b, uv);
    // 2) two-hop gather
    gather_kernel<<<B, 64, 0, stream>>>(v, adj_ent, adj_rel, ents1, rels0, ents2, rels1);

    // 3) iter0 hop1: T1[b*8+n] = ent(ents1) + softmax(uv . rel(rels1)) . ent(ents2)
    agg_kernel<<<B * KNBR, BDIM, 0, stream>>>(uv, ent_table, ents1, ent_table, ents2,
                                              rel_table, rels1, T1, 3);
    //    iter0 hop0: T0[b] = ent(v) + softmax(uv . rel(rels0)) . ent(ents1)
    agg_kernel<<<B, BDIM, 0, stream>>>(uv, ent_table, v, ent_table, ents1,
                                       rel_table, rels0, T0, 0);

    // 4) WMMA GEMMs (relu)
    {
        int rows = B * KNBR;                 // 65536 -> 4096 tiles -> 512 blocks
        int blocks = (rows / 16 + 7) / 8;
        gemm64_wmma_kernel<<<blocks, 256, 0, stream>>>(T1, agg_w, agg_b, ev1, rows, 0);
    }
    {
        int rows = B;                        // 8192 -> 512 tiles -> 64 blocks
        int blocks = (rows / 16 + 7) / 8;
        gemm64_wmma_kernel<<<blocks, 256, 0, stream>>>(T0, agg_w, agg_b, ev0, rows, 0);
    }

    // 5) iter1 hop0: T2[b] = ev0[b] + softmax(uv . rel(rels0)) . ev1[b*8+k]
    agg_kernel<<<B, BDIM, 0, stream>>>(uv, ev0, nullptr, ev1, nullptr,
                                       rel_table, rels0, T2, 0);
    // 6) final GEMM (tanh) -> item embedding
    {
        int rows = B;
        int blocks = (rows / 16 + 7) / 8;
        gemm64_wmma_kernel<<<blocks, 256, 0, stream>>>(T2, agg_w, agg_b, item, rows, 1);
    }

    // 7) item conv + dot + sigmoid
    final_kernel<<<B, BDIM, 0, stream>>>(itemadj, usr_table, item, uv,
                                         conv_w, conv_b, out);
}